// PanSolver2_83519934038070
// MI455X (gfx1250) — compile-verified
//
#include <hip/hip_runtime.h>
#include <math.h>

// ---------------------------------------------------------------------------
// PanSolver2 on MI455X (gfx1250): fp32 WMMA (V_WMMA_F32_16X16X4_F32) GEMMs.
// ~172 GFLOP fp32 per call, ~30MB working set (L2-resident) -> compute bound.
// GEMM: 256x128 block tile, 8 waves, wave tile 32x128 (16 wmma : ~5 ds_2addr
// loads per K-step). A and B staged in LDS with [row][k] stride-36 layout so
// every fragment is one conflict-free ds_load_b64. Global->LDS staging is
// software-pipelined through registers (prefetch next K-chunk during WMMA).
// ---------------------------------------------------------------------------

typedef __attribute__((ext_vector_type(8))) float v8f;
typedef __attribute__((ext_vector_type(2))) float v2f;

#define WMMA_F32(a, b, c) \
  __builtin_amdgcn_wmma_f32_16x16x4_f32(false, (a), false, (b), (short)0, (c), false, false)

// problem sizes
#define BB   128
#define DD   512
#define HH   1024
#define NN   32
#define TT   16
#define MM   63     // 2N-1 simpson points
#define MPAD 64     // padded m dimension
#define RR   (BB * MPAD)  // 8192 rows for the MLP GEMMs

// workspace layout (float offsets)
#define PHIFT_OFF  0                        // 64*32  Phi_f^T (row m, col n), row 63 = 0
#define PHIINV_OFF 2048                     // 32*32  inv(Phi) row-major [k][n]
#define PHIOUT_OFF 3072                     // 32*16  Phi_out [n][t]
#define TTRUE_OFF  3584                     // 64     t_true (index 63 = 0)
#define WID_OFF    3648                     // 32     simpson widths (index 31 = 0)
#define A2_OFF     4096                     // 128*64*512  approx in [b][m][d]
#define H_OFF      (A2_OFF + 4194304)       // 8192*1024   hidden
#define F_OFF      (H_OFF + 8388608)        // 8192*512    f(approx) in [b][m][d]
#define B0_OFF     (F_OFF + 4194304)        // 128*512*32  coeff ping
#define B1_OFF     (B0_OFF + 2097152)       // 128*512*32  coeff pong

#define PI_F 3.14159265358979323846f

// ---------------------------------------------------------------------------
// Setup: Chebyshev grids, Phi_f^T, Gauss-Jordan inverse of Phi, Phi_out.
// ---------------------------------------------------------------------------
__global__ __launch_bounds__(64) void setup_kernel(const float* __restrict__ t_span,
                                                   float* __restrict__ ws) {
  __shared__ float tcheb[NN];
  __shared__ float ts[64];
  __shared__ float aug[32][65];
  __shared__ float fac[32];
  const int tid = threadIdx.x;
  const float t0 = t_span[0];
  const float t1 = t_span[TT - 1];

  if (tid < NN)
    tcheb[tid] = -cosf(PI_F * (float)tid / (float)(NN - 1));
  __syncthreads();

  if (tid < MM)
    ts[tid] = (tid & 1) ? 0.5f * (tcheb[tid >> 1] + tcheb[(tid >> 1) + 1])
                        : tcheb[tid >> 1];
  if (tid == 63) ts[63] = 0.0f;
  __syncthreads();

  // t_true (padded), Phi_f^T row per thread
  ws[TTRUE_OFF + tid] = (tid < MM) ? (t0 + 0.5f * (t1 - t0) * (ts[tid] + 1.0f)) : 0.0f;
  {
    float* row = ws + PHIFT_OFF + tid * NN;
    if (tid < MM) {
      float t = ts[tid], p0 = 1.0f, p1 = t;
      row[0] = p0; row[1] = p1;
      for (int n = 2; n < NN; ++n) { float p2 = 2.0f * t * p1 - p0; row[n] = p2; p0 = p1; p1 = p2; }
    } else {
      for (int n = 0; n < NN; ++n) row[n] = 0.0f;
    }
  }
  // simpson widths: (t_true[2j+2]-t_true[2j])/6
  if (tid < NN)
    ws[WID_OFF + tid] = (tid < NN - 1)
        ? (0.5f * (t1 - t0) * (ts[2 * tid + 2] - ts[2 * tid])) / 6.0f
        : 0.0f;

  // Gauss-Jordan inversion of Phi[n][j] = T_n(tcheb[j]) in LDS.
  // Thread `tid` owns column tid of the [32 x 64] augmented matrix.
  if (tid < 32) {
    float t = tcheb[tid], p0 = 1.0f, p1 = t;
    aug[0][tid] = p0; aug[1][tid] = p1;
    for (int n = 2; n < 32; ++n) { float p2 = 2.0f * t * p1 - p0; aug[n][tid] = p2; p0 = p1; p1 = p2; }
  } else {
    for (int i = 0; i < 32; ++i) aug[i][tid] = ((tid - 32) == i) ? 1.0f : 0.0f;
  }
  __syncthreads();
  for (int k = 0; k < 32; ++k) {
    const float piv = aug[k][k];
    __syncthreads();
    aug[k][tid] /= piv;
    __syncthreads();
    if (tid < 32) fac[tid] = (tid == k) ? 0.0f : aug[tid][k];
    __syncthreads();
    for (int i = 0; i < 32; ++i) aug[i][tid] -= fac[i] * aug[k][tid];
    __syncthreads();
  }
  if (tid >= 32) {
    const int n = tid - 32;
    for (int i = 0; i < 32; ++i) ws[PHIINV_OFF + i * 32 + n] = aug[i][tid];
  }
  // Phi_out[n][t] at t_out = -1 + 2*(t_span - t0)/(t1 - t0)
  if (tid < TT) {
    float t = -1.0f + 2.0f * (t_span[tid] - t0) / (t1 - t0);
    float p0 = 1.0f, p1 = t;
    ws[PHIOUT_OFF + 0 * TT + tid] = p0;
    ws[PHIOUT_OFF + 1 * TT + tid] = p1;
    for (int n = 2; n < NN; ++n) {
      float p2 = 2.0f * t * p1 - p0;
      ws[PHIOUT_OFF + n * TT + tid] = p2;
      p0 = p1; p1 = p2;
    }
  }
}

// ---------------------------------------------------------------------------
// approx[b][m][d] = sum_n Bc[b][d][n] * Phi_f[n][m]   (WMMA, K=32)
// A = Phi_f^T (64x32, shared), B = Bc[b] viewed as [n][d], staged as [d][n].
// Block: 256 thr, computes 64(m) x 128(d) per (b, d-tile).
// ---------------------------------------------------------------------------
__global__ __launch_bounds__(256) void coef2grid_kernel(const float* __restrict__ Bc,
                                                        const float* __restrict__ phiFT,
                                                        float* __restrict__ A2) {
  __shared__ __align__(16) float As[64][36];    // [m][n]
  __shared__ __align__(16) float Bs[128][36];   // [d][n]  (column-major B)
  const int tid = threadIdx.x;
  const int b = blockIdx.y;
  const int d0 = blockIdx.x * 128;

  // Phi_f^T 64x32 -> As
  #pragma unroll
  for (int i = 0; i < 2; ++i) {
    int idx = tid + 256 * i;              // float4 index, 512 total
    int row = idx >> 3;
    int kc = (idx & 7) * 4;
    float4 v = *(const float4*)(phiFT + row * NN + kc);
    *(float4*)&As[row][kc] = v;
  }
  // Bc[b, d0+dcol, n4..n4+3] -> Bs[dcol][n4..n4+3]  (contiguous both sides)
  #pragma unroll
  for (int i = 0; i < 4; ++i) {
    int idx = tid + 256 * i;              // float4 index, 1024 total
    int dcol = idx >> 3;
    int n4 = (idx & 7) * 4;
    float4 v = *(const float4*)(Bc + ((size_t)b * DD + d0 + dcol) * NN + n4);
    *(float4*)&Bs[dcol][n4] = v;
  }
  __syncthreads();

  const int wave = tid >> 5, lane = tid & 31;
  const int wrow = wave >> 1;             // 16-row tile index (m)
  const int wcol = (wave & 1) * 64;       // 64-col group (d)
  const int m16 = lane & 15;
  const int khalf = (lane >> 4) * 2;
  const int rbase = 8 * (lane >> 4);

  const float* arow = &As[wrow * 16 + m16][khalf];
  v8f acc[4] = {};
  #pragma unroll
  for (int k0 = 0; k0 < 32; k0 += 4) {
    v2f a = *(const v2f*)(arow + k0);
    #pragma unroll
    for (int ct = 0; ct < 4; ++ct) {
      v2f bf = *(const v2f*)&Bs[wcol + ct * 16 + m16][k0 + khalf];
      acc[ct] = WMMA_F32(a, bf, acc[ct]);
    }
  }
  #pragma unroll
  for (int ct = 0; ct < 4; ++ct) {
    const int col = d0 + wcol + ct * 16 + m16;
    #pragma unroll
    for (int i = 0; i < 8; ++i) {
      const int row = wrow * 16 + rbase + i;          // m (padded to 64)
      A2[((size_t)b * MPAD + row) * DD + col] = acc[ct][i];
    }
  }
}

// ---------------------------------------------------------------------------
// Generic 256x128-tile WMMA GEMM: C[r][c] = sum_k A[r][k] * Bm[k][c]
//   A: RR x K row-major, Bm: K x ldc row-major, C: RR x ldc.
// If bias != nullptr: C = tanh(acc + t_true[r & 63] * bias[c])   (layer 1)
// else plain store                                               (layer 2)
// Block: 256 thr (8 waves); wave tile 32x128 (acc[2][8]); Kt = 32 in LDS.
// B staged column-major [col][k] so fragments are contiguous ds_load_b64.
// Software pipeline: next K-chunk is prefetched into registers while the
// WMMA loop consumes the current chunk from LDS.
// ---------------------------------------------------------------------------
__global__ __launch_bounds__(256) void gemm128_kernel(const float* __restrict__ A,
                                                      const float* __restrict__ Bm,
                                                      float* __restrict__ C,
                                                      int K, int ldc,
                                                      const float* __restrict__ bias,
                                                      const float* __restrict__ tT_ws) {
  __shared__ __align__(16) float As[256][36];   // [row][k]
  __shared__ __align__(16) float Bs[128][36];   // [col][k]
  __shared__ float tT[64];
  const int tid = threadIdx.x;
  const int r0 = blockIdx.y * 256;
  const int c0 = blockIdx.x * 128;
  if (bias && tid < 64) tT[tid] = tT_ws[tid];

  const int wave = tid >> 5, lane = tid & 31;
  const int m16 = lane & 15;
  const int khalf = (lane >> 4) * 2;
  const int rbase = 8 * (lane >> 4);

  // A staging decomposition: thread -> 8x (row, 4 consecutive k)
  const int arow_ld = tid >> 3;         // 0..31 (+32 per i-step)
  const int akc_ld = (tid & 7) * 4;
  // B staging decomposition: thread -> 4x (col, 4 consecutive k)
  const int bcol = tid & 127;
  const int bk4 = (tid >> 7) * 4;       // 0 or 4; +8 per i-step

  const float* Abase = A + (size_t)(r0 + arow_ld) * K + akc_ld;
  const float* Bbase = Bm + c0 + bcol;

  const float* arow0 = &As[wave * 32 + m16][khalf];
  const float* arow1 = &As[wave * 32 + 16 + m16][khalf];
  const float* brow = &Bs[m16][khalf];

  float4 ra[8];
  float4 rb[4];

  // prologue: fetch chunk 0
  #pragma unroll
  for (int i = 0; i < 8; ++i)
    ra[i] = *(const float4*)(Abase + (size_t)(32 * i) * K);
  #pragma unroll
  for (int i = 0; i < 4; ++i) {
    const int k4 = bk4 + 8 * i;
    rb[i].x = Bbase[(size_t)(k4 + 0) * ldc];
    rb[i].y = Bbase[(size_t)(k4 + 1) * ldc];
    rb[i].z = Bbase[(size_t)(k4 + 2) * ldc];
    rb[i].w = Bbase[(size_t)(k4 + 3) * ldc];
  }

  v8f acc[2][8] = {};
  for (int kk = 0; kk < K; kk += 32) {
    __syncthreads();   // previous chunk fully consumed
    // publish prefetched chunk to LDS
    #pragma unroll
    for (int i = 0; i < 8; ++i)
      *(float4*)&As[arow_ld + 32 * i][akc_ld] = ra[i];
    #pragma unroll
    for (int i = 0; i < 4; ++i)
      *(float4*)&Bs[bcol][bk4 + 8 * i] = rb[i];
    __syncthreads();
    // prefetch next chunk into registers (overlaps with WMMA below)
    if (kk + 32 < K) {
      const int kn = kk + 32;
      #pragma unroll
      for (int i = 0; i < 8; ++i)
        ra[i] = *(const float4*)(Abase + (size_t)(32 * i) * K + kn);
      #pragma unroll
      for (int i = 0; i < 4; ++i) {
        const int k4 = bk4 + 8 * i;
        rb[i].x = Bbase[(size_t)(kn + k4 + 0) * ldc];
        rb[i].y = Bbase[(size_t)(kn + k4 + 1) * ldc];
        rb[i].z = Bbase[(size_t)(kn + k4 + 2) * ldc];
        rb[i].w = Bbase[(size_t)(kn + k4 + 3) * ldc];
      }
    }
    // consume current chunk
    #pragma unroll
    for (int k0 = 0; k0 < 32; k0 += 4) {
      v2f a0 = *(const v2f*)(arow0 + k0);
      v2f a1 = *(const v2f*)(arow1 + k0);
      #pragma unroll
      for (int ct = 0; ct < 8; ++ct) {
        v2f bf = *(const v2f*)(brow + ct * (16 * 36) + k0);
        acc[0][ct] = WMMA_F32(a0, bf, acc[0][ct]);
        acc[1][ct] = WMMA_F32(a1, bf, acc[1][ct]);
      }
    }
  }
  // epilogue
  #pragma unroll
  for (int rt = 0; rt < 2; ++rt) {
    #pragma unroll
    for (int ct = 0; ct < 8; ++ct) {
      const int col = c0 + ct * 16 + m16;
      if (bias) {
        const float bb = bias[col];
        #pragma unroll
        for (int i = 0; i < 8; ++i) {
          const int row = r0 + wave * 32 + rt * 16 + rbase + i;
          C[(size_t)row * ldc + col] = tanhf(acc[rt][ct][i] + tT[row & 63] * bb);
        }
      } else {
        #pragma unroll
        for (int i = 0; i < 8; ++i) {
          const int row = r0 + wave * 32 + rt * 16 + rbase + i;
          C[(size_t)row * ldc + col] = acc[rt][ct][i];
        }
      }
    }
  }
}

// ---------------------------------------------------------------------------
// Simpson quadrature + cumsum + (S + y) @ Phi_inv. One thread per (b,d).
// ---------------------------------------------------------------------------
__global__ __launch_bounds__(256) void simpson_kernel(const float* __restrict__ F,
                                                      const float* __restrict__ y_init,
                                                      const float* __restrict__ phiinv,
                                                      const float* __restrict__ widths,
                                                      float* __restrict__ Bn) {
  __shared__ float Pinv[NN * NN];
  __shared__ float swd[NN];
  const int tid = threadIdx.x;
  #pragma unroll
  for (int i = 0; i < 4; ++i) Pinv[tid + 256 * i] = phiinv[tid + 256 * i];
  if (tid < NN) swd[tid] = widths[tid];
  __syncthreads();

  const int gid = blockIdx.x * 256 + tid;
  const int b = gid >> 9;
  const int d = gid & 511;
  const float y = y_init[(size_t)b * DD + d];
  const size_t base = (size_t)(b * MPAD) * DD + d;

  float S[NN];
  S[0] = 0.0f;
  float accum = 0.0f;
  float fprev = F[base];
  for (int j = 0; j < NN - 1; ++j) {
    const float f1 = F[base + (size_t)(2 * j + 1) * DD];
    const float f2 = F[base + (size_t)(2 * j + 2) * DD];
    accum += swd[j] * (fprev + 4.0f * f1 + f2);
    S[j + 1] = accum;
    fprev = f2;
  }
  float* o = Bn + ((size_t)b * DD + d) * NN;
  #pragma unroll 4
  for (int n = 0; n < NN; ++n) {
    float v = 0.0f;
    #pragma unroll
    for (int k = 0; k < NN; ++k) v += (S[k] + y) * Pinv[k * NN + n];
    o[n] = v;
  }
}

// ---------------------------------------------------------------------------
// Final: out1[t][b][d] = sum_n Bf[b][d][n] * Phi_out[n][t]; out2 = Bf copy.
// ---------------------------------------------------------------------------
__global__ __launch_bounds__(256) void final_kernel(const float* __restrict__ Bf,
                                                    const float* __restrict__ phiout,
                                                    float* __restrict__ out1,
                                                    float* __restrict__ out2) {
  __shared__ float Pout[NN * TT];
  const int tid = threadIdx.x;
  #pragma unroll
  for (int i = 0; i < 2; ++i) Pout[tid + 256 * i] = phiout[tid + 256 * i];
  __syncthreads();

  const int gid = blockIdx.x * 256 + tid;
  const int b = gid >> 9;
  const int d = gid & 511;
  const float* src = Bf + ((size_t)b * DD + d) * NN;
  float c[NN];
  #pragma unroll
  for (int n = 0; n < NN; ++n) c[n] = src[n];

  #pragma unroll
  for (int t = 0; t < TT; ++t) {
    float v = 0.0f;
    #pragma unroll
    for (int n = 0; n < NN; ++n) v += c[n] * Pout[n * TT + t];
    out1[(size_t)t * (BB * DD) + (size_t)b * DD + d] = v;
  }
  float* dst = out2 + ((size_t)b * DD + d) * NN;
  #pragma unroll
  for (int n = 0; n < NN; ++n) dst[n] = c[n];
}

// ---------------------------------------------------------------------------
extern "C" void kernel_launch(void* const* d_in, const int* in_sizes, int n_in,
                              void* d_out, int out_size, void* d_ws, size_t ws_size,
                              hipStream_t stream) {
  (void)in_sizes; (void)n_in; (void)out_size; (void)ws_size;
  const float* t_span = (const float*)d_in[0];
  const float* y_init = (const float*)d_in[1];
  const float* B_init = (const float*)d_in[2];
  const float* W1     = (const float*)d_in[3];
  const float* b1     = (const float*)d_in[4];
  const float* W2     = (const float*)d_in[5];
  float* ws  = (float*)d_ws;
  float* out = (float*)d_out;

  setup_kernel<<<1, 64, 0, stream>>>(t_span, ws);

  const float* phiFT = ws + PHIFT_OFF;
  float* A2 = ws + A2_OFF;
  float* Hb = ws + H_OFF;
  float* Fb = ws + F_OFF;
  float* Bbuf[2] = { ws + B0_OFF, ws + B1_OFF };

  const float* cur = B_init;
  for (int it = 0; it < 10; ++it) {
    float* nxt = Bbuf[it & 1];
    coef2grid_kernel<<<dim3(DD / 128, BB), 256, 0, stream>>>(cur, phiFT, A2);
    gemm128_kernel<<<dim3(HH / 128, RR / 256), 256, 0, stream>>>(
        A2, W1, Hb, DD, HH, b1, ws + TTRUE_OFF);
    gemm128_kernel<<<dim3(DD / 128, RR / 256), 256, 0, stream>>>(
        Hb, W2, Fb, HH, DD, nullptr, nullptr);
    simpson_kernel<<<dim3((BB * DD) / 256), 256, 0, stream>>>(
        Fb, y_init, ws + PHIINV_OFF, ws + WID_OFF, nxt);
    cur = nxt;
  }
  final_kernel<<<dim3((BB * DD) / 256), 256, 0, stream>>>(
      cur, ws + PHIOUT_OFF, out, out + (size_t)TT * BB * DD);
}